// ShoeboxToRIR_29085518529094
// MI455X (gfx1250) — compile-verified
//
#include <hip/hip_runtime.h>
#include <cstdint>

#define SAMPLE_RATE 16000.0f
#define SOUND_SPEED 343.0f
#define RIRLEN      3968
#define PADW        40          // WINDOW_LENGTH 81 -> PAD 40
#define NGRID       31          // indices -15..15
#define NCELL       (31 * 31 * 31)

typedef int v4i __attribute__((ext_vector_type(4)));
typedef __attribute__((address_space(1))) v4i* gv4i_p;   // global int4*
typedef __attribute__((address_space(3))) v4i* lv4i_p;   // LDS int4*

// exact small-integer power by squaring (exponents <= 15)
__device__ __forceinline__ float ipowf(float b, int e) {
    float r = 1.0f;
    while (e) { if (e & 1) r *= b; b *= b; e >>= 1; }
    return r;
}

__global__ __launch_bounds__(1024) void shoebox_rir_kernel(
    const float* __restrict__ inp,   // (B,12)
    float* __restrict__ out,         // B*RIRLEN rir then B toa
    int B)
{
    __shared__ float s_rir[RIRLEN];

    const int b   = blockIdx.x;
    const int tid = threadIdx.x;
    const int nt  = blockDim.x;

    const float* p = inp + b * 12;
    const float rx = p[0], ry = p[1], rz = p[2];
    const float mx = p[3] * rx, my = p[4] * ry, mz = p[5] * rz;
    const float sx = p[6] * rx, sy = p[7] * ry, sz = p[8] * rz;
    // absorption = [aw,aw,aw,aw, p10, p11]*0.84+0.01 ; tr = sqrt(1-a)
    const float trw = sqrtf(1.0f - (p[9]  * 0.84f + 0.01f));  // walls (dims 0,1, lo&hi)
    const float trf = sqrtf(1.0f - (p[10] * 0.84f + 0.01f));  // dim2 lo
    const float trc = sqrtf(1.0f - (p[11] * 0.84f + 0.01f));  // dim2 hi

    for (int t = tid; t < RIRLEN; t += nt) s_rir[t] = 0.0f;
    __syncthreads();

    const float PI_F   = 3.14159265358979323846f;
    const float kDelay = SAMPLE_RATE / SOUND_SPEED;

    for (int cell = tid; cell < NCELL; cell += nt) {
        int qx = cell % NGRID - 15;
        int qy = (cell / NGRID) % NGRID - 15;
        int qz = cell / (NGRID * NGRID) - 15;
        int ax = abs(qx), ay = abs(qy), az = abs(qz);
        if (ax + ay + az > 15) continue;

        // image location: odd index -> d*(q+1)-s ; even -> d*q+s
        float ix = (qx & 1) ? (rx * (float)(qx + 1) - sx) : (rx * (float)qx + sx);
        float iy = (qy & 1) ? (ry * (float)(qy + 1) - sy) : (ry * (float)qy + sy);
        float iz = (qz & 1) ? (rz * (float)(qz + 1) - sz) : (rz * (float)qz + sz);

        // exponents: |floor(q/2)| + |floor((q+1)/2)| == |q| for the wall dims
        int lo_z = abs(qz >> 1), hi_z = abs((qz + 1) >> 1);
        float att = ipowf(trw, ax + ay) * ipowf(trf, lo_z) * ipowf(trc, hi_z);

        float dx = ix - mx, dy = iy - my, dz = iz - mz;
        float dist    = sqrtf(dx * dx + dy * dy + dz * dz);
        float amp     = att / dist;
        float delay   = dist * kDelay;
        float delay_i = ceilf(delay);
        float frac    = delay_i - delay;          // [0,1), exact (Sterbenz)
        int   base    = (int)delay_i - PADW;
        if (base >= RIRLEN) continue;             // whole window past the RIR

        // sin(pi*(frac + k)) = (-1)^k * sin(pi*frac); k = n - 40 -> sign = (-1)^n
        float s0  = sinf(PI_F * frac);
        float sgn = 1.0f;
        for (int n = 0; n <= 2 * PADW; ++n) {
            int t = base + n;
            if ((unsigned)t < (unsigned)RIRLEN) {
                float x = frac + (float)(n - PADW);
                float w = (fabsf(x) <= (float)PADW)
                              ? (0.5f + 0.5f * cosf(x * (PI_F / (float)PADW)))
                              : 0.0f;
                float sc = (x == 0.0f) ? 1.0f : (sgn * s0) / (PI_F * x);
                float v  = amp * sc * w;
                if (v != 0.0f) atomicAdd(&s_rir[t], v);   // ds_add_f32
            }
            sgn = -sgn;
        }
    }

    if (tid == 0) {
        float ddx = mx - sx, ddy = my - sy, ddz = mz - sz;
        out[(size_t)B * RIRLEN + b] =
            (float)PADW + SAMPLE_RATE * sqrtf(ddx * ddx + ddy * ddy + ddz * ddz) / SOUND_SPEED;
    }

    __syncthreads();   // all ds_add_f32 complete before writeback

    float* orow = out + (size_t)b * RIRLEN;
#if defined(__gfx1250__) && __has_builtin(__builtin_amdgcn_global_store_async_from_lds_b128)
    for (int v = tid; v < RIRLEN / 4; v += nt) {
        gv4i_p g = (gv4i_p)(uintptr_t)(orow + 4 * v);
        lv4i_p l = (lv4i_p)(uint32_t)(uintptr_t)(&s_rir[4 * v]);
        __builtin_amdgcn_global_store_async_from_lds_b128(g, l, 0, 0);
    }
#if __has_builtin(__builtin_amdgcn_s_wait_asynccnt)
    __builtin_amdgcn_s_wait_asynccnt(0);
#endif
#else
    for (int v = tid; v < RIRLEN / 4; v += nt) {
        ((float4*)orow)[v] = ((const float4*)s_rir)[v];
    }
#endif
}

extern "C" void kernel_launch(void* const* d_in, const int* in_sizes, int n_in,
                              void* d_out, int out_size, void* d_ws, size_t ws_size,
                              hipStream_t stream) {
    (void)n_in; (void)out_size; (void)d_ws; (void)ws_size;
    const float* inp = (const float*)d_in[0];
    float* out = (float*)d_out;
    const int B = in_sizes[0] / 12;   // (B,12) input
    shoebox_rir_kernel<<<dim3(B), dim3(1024), 0, stream>>>(inp, out, B);
}